// SpacialConv_66168266162365
// MI455X (gfx1250) — compile-verified
//
#include <hip/hip_runtime.h>

// Problem constants (match reference)
#define NN 50000
#define EE 800000
#define DD 128
#define EPSF 1e-7f
#define SLOPE 0.01f
#define LDS_STRIDE 132   // 128 + 4 pad to avoid LDS bank conflicts

typedef __attribute__((ext_vector_type(2))) float v2f;
typedef __attribute__((ext_vector_type(4))) float v4f;
typedef __attribute__((ext_vector_type(8))) float v8f;

// ---------------------------------------------------------------------------
// Kernel 0: zero workspace (s[N][128] followed by deg[N])
// ---------------------------------------------------------------------------
__global__ void zero_kernel(float* __restrict__ p, int n) {
    int i = blockIdx.x * blockDim.x + threadIdx.x;
    int stride = gridDim.x * blockDim.x;
    for (; i < n; i += stride) p[i] = 0.0f;
}

// ---------------------------------------------------------------------------
// Kernel 1: edge phase. One wave32 per edge; lane j owns features 4j..4j+3.
//   e = leaky_relu(coeff @ W_spatial^T + b_spatial)
//   atomicAdd s[dst] += feat[src] * e ; atomicAdd deg[dst] += 1
// Spatial weights (128x3) + bias preloaded into registers per lane.
// ---------------------------------------------------------------------------
__global__ void edge_kernel(const float* __restrict__ feat,
                            const float* __restrict__ pos,
                            const int*   __restrict__ src,
                            const int*   __restrict__ dst,
                            const float* __restrict__ Wsp,   // [128][3]
                            const float* __restrict__ bsp,   // [128]
                            float* __restrict__ s,           // [N][128]
                            float* __restrict__ deg)         // [N]
{
    const int lane  = threadIdx.x & 31;
    const int wave  = (blockIdx.x * blockDim.x + threadIdx.x) >> 5;
    const int nwave = (gridDim.x * blockDim.x) >> 5;
    const int f0    = lane * 4;

    // Per-lane spatial weights for its 4 features (registers)
    float w0[4], w1[4], w2[4], bb[4];
    #pragma unroll
    for (int t = 0; t < 4; ++t) {
        w0[t] = Wsp[(f0 + t) * 3 + 0];
        w1[t] = Wsp[(f0 + t) * 3 + 1];
        w2[t] = Wsp[(f0 + t) * 3 + 2];
        bb[t] = bsp[f0 + t];
    }

    for (int e = wave; e < EE; e += nwave) {
        const int si = src[e];
        const int di = dst[e];
        // rel = pos[dst] - pos[src]  (broadcast loads, coalesced by HW)
        const float rx = pos[di * 3 + 0] - pos[si * 3 + 0];
        const float ry = pos[di * 3 + 1] - pos[si * 3 + 1];
        const float rz = pos[di * 3 + 2] - pos[si * 3 + 2];
        const float inv = 1.0f / (sqrtf(rx * rx + ry * ry + rz * rz) + EPSF);
        const float c0 = (rx + 1.0f) * inv;
        const float c1 = (ry + 1.0f) * inv;
        const float c2 = (rz + 1.0f) * inv;

        const v4f fv = *(const v4f*)(feat + (size_t)si * DD + f0);
        float* sp = s + (size_t)di * DD + f0;
        #pragma unroll
        for (int t = 0; t < 4; ++t) {
            float ev = c0 * w0[t] + c1 * w1[t] + c2 * w2[t] + bb[t];
            ev = (ev > 0.0f) ? ev : ev * SLOPE;          // leaky_relu
            unsafeAtomicAdd(sp + t, fv[t] * ev);          // global_atomic_add_f32
        }
        if (lane == 0) unsafeAtomicAdd(deg + di, 1.0f);
    }
}

// ---------------------------------------------------------------------------
// Kernel 2: node phase with fp32 WMMA (V_WMMA_F32_16X16X4_F32).
// Block = 16 nodes x 256 threads (8 waves). Wave ct computes output cols
// [ct*16, ct*16+16). D = feat_tile @ W_self^T + hmean_tile @ W_neigh^T,
// then + b_self + b_neigh + bias, leaky_relu.
//
// A frag (16x4 f32): lane L holds M=L%16, K = k + (L/16)*2 + {0,1}  -> float2
// B frag (4x16 f32): lane L holds N=L%16, K = k + (L/16)*2 + {0,1}
//                    B[k][n] = W[n][k] (row-major W) -> contiguous float2
// C/D: lane L col N=L%16, VGPR r -> row M = r + (L/16)*8
// ---------------------------------------------------------------------------
__global__ void node_kernel(const float* __restrict__ feat,
                            const float* __restrict__ Wn,   // [128][128] W_neigh
                            const float* __restrict__ bn,
                            const float* __restrict__ Wsf,  // [128][128] W_self
                            const float* __restrict__ bsf,
                            const float* __restrict__ bias,
                            const float* __restrict__ s,    // [N][128]
                            const float* __restrict__ deg,  // [N]
                            float* __restrict__ out)        // [N][128]
{
    __shared__ float Af[16 * LDS_STRIDE];   // feat tile
    __shared__ float Am[16 * LDS_STRIDE];   // h_mean tile

    const int nodeBase = blockIdx.x * 16;
    const int tid = threadIdx.x;

    // Cooperative stage: 2048 elements each, 8 per thread
    for (int i = tid; i < 16 * DD; i += 256) {
        const int m = i >> 7;           // node row in tile
        const int j = i & 127;          // feature
        const int node = nodeBase + m;
        const float d = fmaxf(deg[node], 1.0f);
        Af[m * LDS_STRIDE + j] = feat[(size_t)node * DD + j];
        Am[m * LDS_STRIDE + j] = s[(size_t)node * DD + j] / d;
    }
    __syncthreads();

    const int wave = tid >> 5;          // col tile 0..7
    const int lane = tid & 31;
    const int nloc = lane & 15;
    const int nglb = wave * 16 + nloc;  // output feature column
    const int khalf = (lane >> 4) * 2;  // K sub-offset for this lane half

    v8f c = {};
    // feat @ W_self^T
    #pragma unroll 4
    for (int k = 0; k < DD; k += 4) {
        v2f a = *(const v2f*)(&Af[nloc * LDS_STRIDE + k + khalf]);
        v2f b = *(const v2f*)(&Wsf[(size_t)nglb * DD + k + khalf]);
        c = __builtin_amdgcn_wmma_f32_16x16x4_f32(false, a, false, b,
                                                  (short)0, c, false, false);
    }
    // h_mean @ W_neigh^T (accumulate into same C)
    #pragma unroll 4
    for (int k = 0; k < DD; k += 4) {
        v2f a = *(const v2f*)(&Am[nloc * LDS_STRIDE + k + khalf]);
        v2f b = *(const v2f*)(&Wn[(size_t)nglb * DD + k + khalf]);
        c = __builtin_amdgcn_wmma_f32_16x16x4_f32(false, a, false, b,
                                                  (short)0, c, false, false);
    }

    const float badd = bsf[nglb] + bn[nglb] + bias[nglb];
    const int rowBase = (lane >> 4) * 8;
    #pragma unroll
    for (int r = 0; r < 8; ++r) {
        const int m = rowBase + r;
        float v = c[r] + badd;
        v = (v > 0.0f) ? v : v * SLOPE;  // final leaky_relu
        out[(size_t)(nodeBase + m) * DD + nglb] = v;
    }
}

// ---------------------------------------------------------------------------
extern "C" void kernel_launch(void* const* d_in, const int* in_sizes, int n_in,
                              void* d_out, int out_size, void* d_ws, size_t ws_size,
                              hipStream_t stream) {
    const float* feat      = (const float*)d_in[0];
    const float* pos       = (const float*)d_in[1];
    const int*   src       = (const int*)  d_in[2];
    const int*   dst       = (const int*)  d_in[3];
    const float* W_spatial = (const float*)d_in[4];
    const float* b_spatial = (const float*)d_in[5];
    const float* W_neigh   = (const float*)d_in[6];
    const float* b_neigh   = (const float*)d_in[7];
    const float* W_self    = (const float*)d_in[8];
    const float* b_self    = (const float*)d_in[9];
    const float* bias      = (const float*)d_in[10];
    float* out = (float*)d_out;

    // Workspace: s[N][128] then deg[N]  (needs (N*128 + N)*4 = ~25.8 MB)
    float* s   = (float*)d_ws;
    float* deg = s + (size_t)NN * DD;

    zero_kernel<<<512, 256, 0, stream>>>(s, NN * DD + NN);
    edge_kernel<<<2048, 256, 0, stream>>>(feat, pos, src, dst,
                                          W_spatial, b_spatial, s, deg);
    node_kernel<<<NN / 16, 256, 0, stream>>>(feat, W_neigh, b_neigh,
                                             W_self, b_self, bias,
                                             s, deg, out);
}